// QLayer_65481071401566
// MI455X (gfx1250) — compile-verified
//
#include <hip/hip_runtime.h>

// CDNA5 / gfx1250: wave32, WMMA, 320KB LDS per WGP.
// One workgroup (256 threads = 8 wave32) per batch element; full 2^14 complex64
// state resident in LDS. Gate-fusion plan (LDS-bandwidth roofline):
//  - encoding RY folded into layer-0 per-wire gates (per-batch fused unitaries)
//  - rotations on wire groups {0-3},{4-7},{8-11} -> 16x16 complex unitaries
//    applied with chained V_WMMA_F32_16X16X4_F32 (full fp32 precision)
//  - CNOT(0,1..2,3) folded into layer-0 group-0 unitary (row remap)
//  - CNOT runs on windows {3-6},{6-9},{9-12} -> one permute sweep each
//  - wires 12,13 rotations -> one fused 4x4 two-qubit butterfly
//  - last layer's whole CNOT chain folded into the <Z> measurement indexing

typedef __attribute__((ext_vector_type(2))) float v2f;
typedef __attribute__((ext_vector_type(8))) float v8f;

#define N_WIRES 14
#define N_LAYERS 2
#define BATCH 128
#define DIM (1 << N_WIRES)      // 16384 amplitudes
#define THREADS 256             // 8 wave32

#define ST_ELEMS DIM                          // float2
#define U_ELEMS (N_LAYERS * 3 * 256)          // float2: 6 fused 16x16 unitaries
#define G_ELEMS (N_LAYERS * N_WIRES * 4)      // float2: per-wire 2x2 gates
#define ZP_ELEMS (8 * N_WIRES)                // float: per-wave partial sums
#define SMEM_BYTES ((size_t)(ST_ELEMS + U_ELEMS + G_ELEMS) * sizeof(float2) + \
                    (size_t)ZP_ELEMS * sizeof(float))

__device__ __forceinline__ float2 cmul(float2 a, float2 b) {
    return make_float2(a.x * b.x - a.y * b.y, a.x * b.y + a.y * b.x);
}
__device__ __forceinline__ float2 cadd(float2 a, float2 b) {
    return make_float2(a.x + b.x, a.y + b.y);
}
__device__ __forceinline__ float2 scl(float s, float2 a) {
    return make_float2(s * a.x, s * a.y);
}

// CNOT(control wire cw, target wire tw): swap amplitudes with c=1, t=0 <-> t=1.
__device__ __forceinline__ void apply_cnot(float2* st, int cw, int tw) {
    const int cb = 13 - cw, tb = 13 - tw;
    const int lo = cb < tb ? cb : tb;
    const int hi = cb < tb ? tb : cb;
    for (int q = threadIdx.x; q < (DIM >> 2); q += THREADS) {  // 4096/256: uniform
        int i = ((q >> lo) << (lo + 1)) | (q & ((1 << lo) - 1));
        i = ((i >> hi) << (hi + 1)) | (i & ((1 << hi) - 1));
        i |= (1 << cb);               // control = 1, target = 0
        int j = i | (1 << tb);
        float2 a = st[i], b = st[j];
        st[i] = b;
        st[j] = a;
    }
    __syncthreads();
}

// One sweep applying C(a,a+1),C(a+1,a+2),C(a+2,a+3) on a 4-wire window whose
// bits occupy state bits [sh+3 : sh] (window bit 3 = wire a). Scatter form:
// new[sigma(m)] = old[m], sigma = C(a+2,a+3) o C(a+1,a+2) o C(a,a+1).
__device__ __forceinline__ void apply_window_perm(float2* st, int sh) {
    const int lowm = (1 << sh) - 1;
    for (int c = threadIdx.x; c < 1024; c += THREADS) {   // 1024/256: uniform
        int base = ((c >> sh) << (sh + 4)) | (c & lowm);
        float2 v[16];
#pragma unroll
        for (int m = 0; m < 16; ++m) v[m] = st[base + (m << sh)];
#pragma unroll
        for (int m = 0; m < 16; ++m) {
            int t = m ^ ((((m >> 3) & 1)) << 2);  // C(a,a+1):  bit3 -> bit2
            t ^= ((t >> 2) & 1) << 1;             // C(a+1,a+2): bit2 -> bit1
            t ^= (t >> 1) & 1;                    // C(a+2,a+3): bit1 -> bit0
            st[base + (t << sh)] = v[m];
        }
    }
    __syncthreads();
}

// Fused two-qubit gate V = A (x) B on wires 12,13 (state bits 1,0): one sweep.
__device__ __forceinline__ void apply2q(float2* st, const float2 (&A)[4],
                                        const float2 (&B)[4]) {
    float2 V[4][4];
#pragma unroll
    for (int r = 0; r < 4; ++r)
#pragma unroll
        for (int c = 0; c < 4; ++c)
            V[r][c] = cmul(A[(r >> 1) * 2 + (c >> 1)], B[(r & 1) * 2 + (c & 1)]);
    for (int q = threadIdx.x; q < (DIM >> 2); q += THREADS) {  // uniform
        int i = q << 2;
        float2 s[4];
#pragma unroll
        for (int c = 0; c < 4; ++c) s[c] = st[i + c];
#pragma unroll
        for (int r = 0; r < 4; ++r) {
            float2 acc = cmul(V[r][0], s[0]);
#pragma unroll
            for (int c = 1; c < 4; ++c) acc = cadd(acc, cmul(V[r][c], s[c]));
            st[i + r] = acc;
        }
    }
    __syncthreads();
}

// Apply a fused 16x16 complex unitary Ug to the 4 wires at state bits
// [sh+3 : sh]. State = B-matrix (16 rows x 1024 cols); per 16-col tile:
// D = U x S via chained f32 WMMA (K=16 as 4 steps of K=4).
// Dr = Ur*Sr + (-Ui)*Si ; Di = Ur*Si + Ui*Sr (f32 WMMA NEG is C-only ->
// pre-negate Ui). Each tile reads/writes its own slab -> in-place, EXEC full.
__device__ __forceinline__ void apply_group_wmma(float2* st, const float2* Ug, int sh) {
    const int lane = threadIdx.x & 31;
    const int wave = threadIdx.x >> 5;
    const int half = lane >> 4;
    const int nidx = lane & 15;

    // A 16x4 f32 fragments: lane = M; V0 = K(kk+2*half), V1 = K+1.
    v2f ur[4], ui[4], nui[4];
#pragma unroll
    for (int kki = 0; kki < 4; ++kki) {
        int k0 = kki * 4 + 2 * half;
        float2 u0 = Ug[nidx * 16 + k0];
        float2 u1 = Ug[nidx * 16 + k0 + 1];
        ur[kki]  = (v2f){u0.x, u1.x};
        ui[kki]  = (v2f){u0.y, u1.y};
        nui[kki] = (v2f){-u0.y, -u1.y};
    }

    const int lowm = (1 << sh) - 1;
    for (int t = 0; t < 8; ++t) {                // 64 tiles / 8 waves
        int c = (wave * 8 + t) * 16 + nidx;
        int base = ((c >> sh) << (sh + 4)) | (c & lowm);
        v8f dr = {};
        v8f di = {};
#pragma unroll
        for (int kki = 0; kki < 4; ++kki) {
            int m0 = kki * 4 + 2 * half;          // B 4x16: rows kk+2*half, +1
            float2 s0 = st[base + (m0 << sh)];
            float2 s1 = st[base + ((m0 + 1) << sh)];
            v2f br = (v2f){s0.x, s1.x};
            v2f bi = (v2f){s0.y, s1.y};
            dr = __builtin_amdgcn_wmma_f32_16x16x4_f32(false, ur[kki],  false, br, (short)0, dr, false, false);
            dr = __builtin_amdgcn_wmma_f32_16x16x4_f32(false, nui[kki], false, bi, (short)0, dr, false, false);
            di = __builtin_amdgcn_wmma_f32_16x16x4_f32(false, ur[kki],  false, bi, (short)0, di, false, false);
            di = __builtin_amdgcn_wmma_f32_16x16x4_f32(false, ui[kki],  false, br, (short)0, di, false, false);
        }
#pragma unroll
        for (int v = 0; v < 8; ++v)               // D: rows v+8*half, col lane&15
            st[base + ((v + 8 * half) << sh)] = make_float2(dr[v], di[v]);
    }
    __syncthreads();
}

__global__ __launch_bounds__(THREADS) void qsim_kernel(
    const float* __restrict__ x,       // (BATCH, 14)
    const float* __restrict__ params,  // (2, 14, 3)
    float* __restrict__ out) {         // (BATCH, 14)
    extern __shared__ unsigned char smem_raw[];
    float2* st    = (float2*)smem_raw;          // 16384 complex amplitudes
    float2* Umat  = st + ST_ELEMS;              // 6 fused 16x16 unitaries
    float2* Gmat  = Umat + U_ELEMS;             // 28 per-wire 2x2 gates
    float*  zpart = (float*)(Gmat + G_ELEMS);   // 8x14 reduction scratch

    const int tid = threadIdx.x;
    const int b = blockIdx.x;

    // |0...0>
    for (int k = tid; k < DIM; k += THREADS) st[k] = make_float2(0.f, 0.f);

    // Per-(layer,wire) gate G = RX(a2)*RZ(a1)*RY(a0); layer 0 additionally
    // absorbs the data-encoding RY(x[b,w]) on the right: G <- G * RY(x).
    if (tid < N_LAYERS * N_WIRES) {
        int l = tid / N_WIRES, w = tid % N_WIRES;
        const float* pp = params + tid * 3;
        float a0 = pp[0], a1 = pp[1], a2 = pp[2];
        float cy = cosf(0.5f * a0), sy = sinf(0.5f * a0);
        float2 em = make_float2(cosf(0.5f * a1), -sinf(0.5f * a1));  // e^{-i a1/2}
        float2 ep = make_float2(em.x, -em.y);                        // e^{+i a1/2}
        float2 m00 = scl(cy, em),  m01 = scl(-sy, em);               // RZ*RY
        float2 m10 = scl(sy, ep),  m11 = scl(cy, ep);
        float cx = cosf(0.5f * a2), sx = sinf(0.5f * a2);
        float2 rxo = make_float2(0.f, -sx);                          // -i*sin
        float2 g00 = cadd(scl(cx, m00), cmul(rxo, m10));
        float2 g01 = cadd(scl(cx, m01), cmul(rxo, m11));
        float2 g10 = cadd(cmul(rxo, m00), scl(cx, m10));
        float2 g11 = cadd(cmul(rxo, m01), scl(cx, m11));
        if (l == 0) {  // fold encoding RY: columns -> [G*(c,s), G*(-s,c)]
            float ce = cosf(0.5f * x[b * N_WIRES + w]);
            float se = sinf(0.5f * x[b * N_WIRES + w]);
            float2 n00 = cadd(scl(ce, g00), scl(se, g01));
            float2 n01 = cadd(scl(-se, g00), scl(ce, g01));
            float2 n10 = cadd(scl(ce, g10), scl(se, g11));
            float2 n11 = cadd(scl(-se, g10), scl(ce, g11));
            g00 = n00; g01 = n01; g10 = n10; g11 = n11;
        }
        float2* G = Gmat + tid * 4;
        G[0] = g00; G[1] = g01; G[2] = g10; G[3] = g11;
    }
    if (tid == 0) st[0] = make_float2(1.f, 0.f);
    __syncthreads();

    // Fused 16x16 unitaries, groups {0..3},{4..7},{8..11} per layer:
    // U[i][j] = prod_t G[4g+t][bit_{3-t}(i)][bit_{3-t}(j)]  (Kronecker).
    // Layer 0 / group 0 folds CNOT(0,1),CNOT(1,2),CNOT(2,3): row remap
    // U'[i][j] = U[a(b(d(i)))][j].
    for (int e = tid; e < N_LAYERS * 3 * 256; e += THREADS) {
        int lg = e >> 8;                 // 0..5 (uniform per iteration)
        int l = lg / 3, g = lg % 3;
        int i = (e >> 4) & 15, j = e & 15;
        int ii = i;
        if (l == 0 && g == 0) {
            ii ^= (ii >> 1) & 1;           // d = C(2,3): bit1 -> bit0
            ii ^= ((ii >> 2) & 1) << 1;    // b = C(1,2): bit2 -> bit1
            ii ^= ((ii >> 3) & 1) << 2;    // a = C(0,1): bit3 -> bit2
        }
        float2 prod = make_float2(1.f, 0.f);
#pragma unroll
        for (int t = 0; t < 4; ++t) {
            int bi_ = (ii >> (3 - t)) & 1;
            int bj  = (j >> (3 - t)) & 1;
            prod = cmul(prod, Gmat[(l * N_WIRES + 4 * g + t) * 4 + bi_ * 2 + bj]);
        }
        Umat[e] = prod;
    }
    __syncthreads();

    for (int l = 0; l < N_LAYERS; ++l) {
        // Rotations (+ folded pieces) on wires 0..11 via WMMA.
        for (int g = 0; g < 3; ++g)
            apply_group_wmma(st, Umat + (l * 3 + g) * 256, 10 - 4 * g);
        // Wires 12,13: one fused 4x4 two-qubit sweep.
        float2 A[4], B[4];
#pragma unroll
        for (int k = 0; k < 4; ++k) {
            A[k] = Gmat[(l * N_WIRES + 12) * 4 + k];
            B[k] = Gmat[(l * N_WIRES + 13) * 4 + k];
        }
        apply2q(st, A, B);
        if (l == 0) {
            // Remaining chain of layer 0 (C(0,1..2,3) already in U0):
            apply_window_perm(st, 7);   // C(3,4),C(4,5),C(5,6)    bits 10..7
            apply_window_perm(st, 4);   // C(6,7),C(7,8),C(8,9)    bits 7..4
            apply_window_perm(st, 1);   // C(9,10),C(10,11),C(11,12) bits 4..1
            apply_cnot(st, 12, 13);
            apply_cnot(st, 13, 0);
        }
        // Layer 1's whole CNOT chain is folded into the measurement below.
    }

    // <Z_w> = sum_m |s_m|^2 * sign_w(sigma(m)), sigma = last layer's chain.
    float z[N_WIRES];
#pragma unroll
    for (int w = 0; w < N_WIRES; ++w) z[w] = 0.f;
    for (int k = tid; k < DIM; k += THREADS) {
        float2 a = st[k];
        float p = a.x * a.x + a.y * a.y;
        int kp = k;
#pragma unroll
        for (int w = 0; w < 13; ++w)               // C(w,w+1) sequentially
            kp ^= ((kp >> (13 - w)) & 1) << (12 - w);
        kp ^= (kp & 1) << 13;                      // C(13,0)
#pragma unroll
        for (int w = 0; w < N_WIRES; ++w)
            z[w] += ((kp >> (13 - w)) & 1) ? -p : p;
    }
    const int lane = tid & 31, wave = tid >> 5;
#pragma unroll
    for (int w = 0; w < N_WIRES; ++w) {
        float v = z[w];
        for (int off = 16; off > 0; off >>= 1) v += __shfl_down(v, off, 32);
        if (lane == 0) zpart[wave * N_WIRES + w] = v;
    }
    __syncthreads();
    if (tid < N_WIRES) {
        float s = 0.f;
#pragma unroll
        for (int wv = 0; wv < 8; ++wv) s += zpart[wv * N_WIRES + tid];
        out[b * N_WIRES + tid] = s;
    }
}

extern "C" void kernel_launch(void* const* d_in, const int* in_sizes, int n_in,
                              void* d_out, int out_size, void* d_ws, size_t ws_size,
                              hipStream_t stream) {
    (void)in_sizes; (void)n_in; (void)out_size; (void)d_ws; (void)ws_size;
    const float* x = (const float*)d_in[0];
    const float* params = (const float*)d_in[1];
    float* out = (float*)d_out;
    hipFuncSetAttribute((const void*)qsim_kernel,
                        hipFuncAttributeMaxDynamicSharedMemorySize,
                        (int)SMEM_BYTES);
    qsim_kernel<<<BATCH, THREADS, SMEM_BYTES, stream>>>(x, params, out);
}